// SCLLoss_65060164600439
// MI455X (gfx1250) — compile-verified
//
#include <hip/hip_runtime.h>
#include <hip/hip_bf16.h>
#include <math.h>

typedef __attribute__((ext_vector_type(2))) float v2f;
typedef __attribute__((ext_vector_type(8))) float v8f;

#define C_DIM 128
#define KSTEPS (C_DIM / 4)           // 32 wmma f32 16x16x4 steps over K=128
#define CH 8                          // column chunks (deterministic partials)
#define T_INV 10.0f                   // 1/T
#define K2E 14.426950408889634f       // log2(e)/T

// ---------------- Kernel 1: row L2-normalize (one wave32 per row) ----------------
__global__ void __launch_bounds__(256) scl_normalize(const float* __restrict__ X,
                                                     float* __restrict__ Xn, int N) {
    const int wave = threadIdx.x >> 5;
    const int lane = threadIdx.x & 31;
    const int r = blockIdx.x * 8 + wave;
    if (r >= N) return;
    float4 v = ((const float4*)(X + (size_t)r * C_DIM))[lane];   // 4 floats/lane * 32 = 128
    float ss = v.x * v.x + v.y * v.y + v.z * v.z + v.w * v.w;
#pragma unroll
    for (int m = 1; m < 32; m <<= 1) ss += __shfl_xor(ss, m, 32);
    const float inv = 1.0f / fmaxf(sqrtf(ss), 1e-12f);
    v.x *= inv; v.y *= inv; v.z *= inv; v.w *= inv;
    ((float4*)(Xn + (size_t)r * C_DIM))[lane] = v;
}

// ---------------- Kernel 2: fused Gram tile (f32 WMMA) + exp/mask reduction ------
// One wave per block. blockIdx.x = row tile (16 rows), blockIdx.y = column chunk.
__global__ void __launch_bounds__(32) scl_gram(const float* __restrict__ Xn,
                                               const int* __restrict__ y,
                                               float* __restrict__ denomP,
                                               float* __restrict__ PP, int N) {
    const int lane = threadIdx.x;
    const int half = lane >> 4;        // K-pair select for A/B fragments
    const int lrow = lane & 15;        // A: M index; B/D: N index
    const int i0 = blockIdx.x * 16;
    const int chunk = blockIdx.y;
    const int colsPerChunk = N / CH;
    const int tilesPerChunk = colsPerChunk / 16;
    const int j0base = chunk * colsPerChunk;

    // A fragments for all 32 K-steps, loaded once, reused for every column tile.
    // Layout (ISA 7.12.2, 32-bit A 16x4): lane 0-15 -> K {0,1}; lane 16-31 -> K {2,3}.
    const float* Arow = Xn + (size_t)(i0 + lrow) * C_DIM + 2 * half;
    v2f a[KSTEPS];
#pragma unroll
    for (int c = 0; c < KSTEPS; ++c) a[c] = *(const v2f*)(Arow + 4 * c);

    // Class labels of the 8 D rows this lane accumulates (M = v + 8*half).
    int yrow[8];
#pragma unroll
    for (int v = 0; v < 8; ++v) yrow[v] = y[i0 + v + 8 * half];

    float dAcc[8], pAcc[8];
#pragma unroll
    for (int v = 0; v < 8; ++v) { dAcc[v] = 0.0f; pAcc[v] = 0.0f; }

    for (int jt = 0; jt < tilesPerChunk; ++jt) {
        const int j0 = j0base + jt * 16;
        const float* Brow = Xn + (size_t)(j0 + lrow) * C_DIM + 2 * half;
        v8f cf = {};
#pragma unroll
        for (int c = 0; c < KSTEPS; ++c) {
            v2f b = *(const v2f*)(Brow + 4 * c);
            // D = A x B + C, full fp32 (matches reference precision)
            cf = __builtin_amdgcn_wmma_f32_16x16x4_f32(
                false, a[c], false, b, (short)0, cf, false, false);
        }
        const int gj = j0 + lrow;      // column handled by this lane (D layout: N = lane&15)
        const int yj = y[gj];
#pragma unroll
        for (int v = 0; v < 8; ++v) {
            const float s = cf[v];
            const int gi = i0 + v + 8 * half;   // D layout: M = v + 8*(lane>>4)
            const float e = exp2f(s * K2E);     // exp(s/T)
            if (gi != gj) {
                dAcc[v] += e;                   // denom accumulation (self excluded)
                if (yj == yrow[v]) pAcc[v] += s;// same-class dot-product sum
            }
        }
    }

    // Reduce across the 16 lanes sharing each D row (butterfly stays in half-group),
    // then write deterministic per-chunk partials (one writer per slot, no atomics).
#pragma unroll
    for (int v = 0; v < 8; ++v) {
        float d = dAcc[v], p = pAcc[v];
#pragma unroll
        for (int m = 1; m < 16; m <<= 1) {
            d += __shfl_xor(d, m, 32);
            p += __shfl_xor(p, m, 32);
        }
        if (lrow == 0) {
            const int r = i0 + v + 8 * half;
            denomP[(size_t)chunk * N + r] = d;
            PP[(size_t)chunk * N + r] = p;
        }
    }
}

// ---------------- Kernel 3: finalize (histogram -> per-row loss -> scalar) -------
__global__ void __launch_bounds__(1024) scl_finalize(const float* __restrict__ denomP,
                                                     const float* __restrict__ PP,
                                                     const int* __restrict__ y,
                                                     float* __restrict__ out, int N) {
    __shared__ int hist[16];
    __shared__ float red[1024];
    const int tid = threadIdx.x;
    if (tid < 16) hist[tid] = 0;
    __syncthreads();
    for (int r = tid; r < N; r += 1024) atomicAdd(&hist[y[r]], 1);
    __syncthreads();

    float acc = 0.0f;
    for (int r = tid; r < N; r += 1024) {
        float d = 0.0f, p = 0.0f;
#pragma unroll
        for (int c = 0; c < CH; ++c) {          // deterministic fixed-order chunk sum
            d += denomP[(size_t)c * N + r];
            p += PP[(size_t)c * N + r];
        }
        const int m = hist[y[r]] - 1;           // same-class count minus self
        const float cnt = (m > 0) ? (float)m : 1.0f;
        float row = -(p * T_INV - (float)m * logf(d)) / cnt;
        if (isnan(row) || isinf(row)) row = 0.0f;
        acc += row;
    }
    red[tid] = acc;
    __syncthreads();
    for (int s = 512; s > 0; s >>= 1) {
        if (tid < s) red[tid] += red[tid + s];
        __syncthreads();
    }
    if (tid == 0) out[0] = red[0];
}

// ---------------- Launch ---------------------------------------------------------
extern "C" void kernel_launch(void* const* d_in, const int* in_sizes, int n_in,
                              void* d_out, int out_size, void* d_ws, size_t ws_size,
                              hipStream_t stream) {
    const float* X = (const float*)d_in[0];
    const int* y = (const int*)d_in[1];
    float* out = (float*)d_out;
    const int N = in_sizes[1];                  // 8192

    float* Xn = (float*)d_ws;                   // N*C floats       (4 MB)
    float* denomP = Xn + (size_t)N * C_DIM;     // CH*N floats      (256 KB)
    float* PP = denomP + (size_t)CH * N;        // CH*N floats      (256 KB)

    scl_normalize<<<dim3((N + 7) / 8), 256, 0, stream>>>(X, Xn, N);
    scl_gram<<<dim3(N / 16, CH), 32, 0, stream>>>(Xn, y, denomP, PP, N);
    scl_finalize<<<1, 1024, 0, stream>>>(denomP, PP, y, out, N);
}